// HungarianMatcher_54786602827871
// MI455X (gfx1250) — compile-verified
//
#include <hip/hip_runtime.h>
#include <stdint.h>

// Problem shape (fixed by the harness reference):
//   pred_logits [16,1000,91] f32, pred_boxes [16,1000,4] f32,
//   tgt_labels  [16,128] i32,     tgt_boxes  [16,128,4] f32
//   out         [16,1000,128] f32
#define NB 16
#define NQ 1000
#define NC 91
#define NT 128
#define TQ 16                       // queries per block tile
#define NQT ((NQ + TQ - 1) / TQ)    // 63 tiles per batch

// ---------------- CDNA5 async global->LDS helpers ----------------
// GV-mode async copy: LDS byte address in a VGPR, 64-bit global addr in a
// VGPR pair. Tracked by ASYNCcnt (CDNA5 ISA).
__device__ __forceinline__ void async_g2l_b128(uint32_t lds_off, const void* g) {
    asm volatile("global_load_async_to_lds_b128 %0, %1, off"
                 :: "v"(lds_off), "v"((uint64_t)(uintptr_t)g) : "memory");
}
__device__ __forceinline__ void async_g2l_b32(uint32_t lds_off, const void* g) {
    asm volatile("global_load_async_to_lds_b32 %0, %1, off"
                 :: "v"(lds_off), "v"((uint64_t)(uintptr_t)g) : "memory");
}
__device__ __forceinline__ void wait_async0() {
#if __has_builtin(__builtin_amdgcn_s_wait_asynccnt)
    __builtin_amdgcn_s_wait_asynccnt(0);
#else
    asm volatile("s_wait_asynccnt 0" ::: "memory");
#endif
}
// Generic LDS pointer = {aperture_hi32, lds_offset}; low 32 bits are the raw
// LDS byte address (CDNA5 ISA aperture table).
__device__ __forceinline__ uint32_t lds_addr(const void* p) {
    return (uint32_t)(uintptr_t)p;
}

__global__ __launch_bounds__(128)
void matcher_cost_kernel(const float* __restrict__ pred_logits,
                         const float* __restrict__ pred_boxes,
                         const int*   __restrict__ tgt_labels,
                         const float* __restrict__ tgt_boxes,
                         float*       __restrict__ out)
{
    __shared__ float s_logit[TQ * NC];   // raw logits tile (16x91)
    __shared__ float s_pbox[TQ * 4];     // pred boxes tile (cx,cy,w,h)

    const int tid  = threadIdx.x;                 // 0..127 == target index t
    const int b    = blockIdx.x / NQT;
    const int q0   = (blockIdx.x % NQT) * TQ;
    const int rows = (NQ - q0 < TQ) ? (NQ - q0) : TQ;

    // ---- async stage: logits tile (contiguous rows*91*4 bytes) ----
    const char*    gl    = (const char*)(pred_logits + ((size_t)b * NQ + q0) * NC);
    const uint32_t lbyt  = (uint32_t)rows * NC * 4u;
    const uint32_t l0    = lds_addr(&s_logit[0]);
    for (uint32_t off = (uint32_t)tid * 16u; off + 16u <= lbyt; off += 128u * 16u)
        async_g2l_b128(l0 + off, gl + off);
    const uint32_t base16 = lbyt & ~15u;          // 4-byte tail (0 for rows in {8,16})
    if (base16 + (uint32_t)tid * 4u < lbyt)
        async_g2l_b32(l0 + base16 + tid * 4u, gl + base16 + tid * 4u);

    // ---- async stage: pred-box tile (rows * 16 bytes) ----
    const char*    gp = (const char*)(pred_boxes + ((size_t)b * NQ + q0) * 4);
    const uint32_t p0 = lds_addr(&s_pbox[0]);
    if (tid < rows)
        async_g2l_b128(p0 + (uint32_t)tid * 16u, gp + (uint32_t)tid * 16u);

    // ---- this thread's target: label + box, kept in registers ----
    const int    lab = tgt_labels[b * NT + tid];
    const float4 tb  = ((const float4*)tgt_boxes)[b * NT + tid];   // cx,cy,w,h
    const float  tx0 = tb.x - 0.5f * tb.z, ty0 = tb.y - 0.5f * tb.w;
    const float  tx1 = tb.x + 0.5f * tb.z, ty1 = tb.y + 0.5f * tb.w;
    const float  t_area = (tx1 - tx0) * (ty1 - ty0);

    wait_async0();          // drain this wave's ASYNCcnt (LDS writes landed)
    __syncthreads();        // make all waves' staged data visible

    float* orow = out + (((size_t)b * NQ + q0) * NT) + tid;

    for (int qi = 0; qi < rows; ++qi) {
        const float pcx = s_pbox[qi * 4 + 0];
        const float pcy = s_pbox[qi * 4 + 1];
        const float pw  = s_pbox[qi * 4 + 2];
        const float ph  = s_pbox[qi * 4 + 3];

        // --- focal class cost: p = sigmoid(logit[q, label_t]) (LDS gather) ---
        const float x   = s_logit[qi * NC + lab];
        const float p   = 1.0f / (1.0f + __expf(-x));
        const float omp = 1.0f - p;
        const float pos = 0.25f * omp * omp * (-__logf(p   + 1e-8f));
        const float neg = 0.75f * p   * p   * (-__logf(omp + 1e-8f));
        const float c_cls = pos - neg;

        // --- L1 on raw cxcywh ---
        const float c_l1 = fabsf(pcx - tb.x) + fabsf(pcy - tb.y)
                         + fabsf(pw  - tb.z) + fabsf(ph  - tb.w);

        // --- GIoU on converted xyxy ---
        const float px0 = pcx - 0.5f * pw, py0 = pcy - 0.5f * ph;
        const float px1 = pcx + 0.5f * pw, py1 = pcy + 0.5f * ph;
        const float p_area = (px1 - px0) * (py1 - py0);
        const float iw = fmaxf(fminf(px1, tx1) - fmaxf(px0, tx0), 0.0f);
        const float ih = fmaxf(fminf(py1, ty1) - fmaxf(py0, ty0), 0.0f);
        const float inter = iw * ih;
        const float uni   = p_area + t_area - inter;
        const float ew = fmaxf(fmaxf(px1, tx1) - fminf(px0, tx0), 0.0f);
        const float eh = fmaxf(fmaxf(py1, ty1) - fminf(py0, ty0), 0.0f);
        const float earea = ew * eh;
        const float giou  = inter / uni - (earea - uni) / earea;

        orow[(size_t)qi * NT] = 5.0f * c_l1 + 2.0f * c_cls - 2.0f * giou;
    }
}

extern "C" void kernel_launch(void* const* d_in, const int* in_sizes, int n_in,
                              void* d_out, int out_size, void* d_ws, size_t ws_size,
                              hipStream_t stream) {
    (void)in_sizes; (void)n_in; (void)out_size; (void)d_ws; (void)ws_size;
    const float* pred_logits = (const float*)d_in[0];
    const float* pred_boxes  = (const float*)d_in[1];
    const int*   tgt_labels  = (const int*)d_in[2];
    const float* tgt_boxes   = (const float*)d_in[3];
    float*       out         = (float*)d_out;

    dim3 grid(NB * NQT);   // 16 * 63 = 1008 blocks
    dim3 block(128);       // 4 wave32 waves; thread == target index
    matcher_cost_kernel<<<grid, block, 0, stream>>>(
        pred_logits, pred_boxes, tgt_labels, tgt_boxes, out);
}